// GNNProcessor_25744033973010
// MI455X (gfx1250) — compile-verified
//
#include <hip/hip_runtime.h>

#define N_NODES 100000
#define N_EDGES 1600000
#define DIM 64

typedef float v2f __attribute__((ext_vector_type(2)));
typedef float v8f __attribute__((ext_vector_type(8)));

// ---------------------------------------------------------------------------
// Zero a float buffer (float4 per thread).
// ---------------------------------------------------------------------------
__global__ void zero_kernel(float* __restrict__ p, int n4) {
    int i = blockIdx.x * blockDim.x + threadIdx.x;
    if (i < n4) {
        ((float4*)p)[i] = make_float4(0.f, 0.f, 0.f, 0.f);
    }
}

// ---------------------------------------------------------------------------
// Edge scatter-add: agg[dst] += feat[src].
// One thread handles one (edge, 4-feature group): float4 gather + 4 HW
// f32 atomics (global_atomic_add_f32, non-returning).
// ---------------------------------------------------------------------------
__global__ void scatter_add_kernel(const float* __restrict__ feat,
                                   const int* __restrict__ src,
                                   const int* __restrict__ dst,
                                   float* __restrict__ agg) {
    long long tid = (long long)blockIdx.x * blockDim.x + threadIdx.x;
    int e  = (int)(tid >> 4);          // edge index
    int f4 = ((int)tid & 15) * 4;      // feature group base
    if (e >= N_EDGES) return;
    int s = src[e];
    int d = dst[e];
    const float4 v = *(const float4*)(feat + (size_t)s * DIM + f4);
    float* p = agg + (size_t)d * DIM + f4;
    unsafeAtomicAdd(p + 0, v.x);
    unsafeAtomicAdd(p + 1, v.y);
    unsafeAtomicAdd(p + 2, v.z);
    unsafeAtomicAdd(p + 3, v.w);
}

// ---------------------------------------------------------------------------
// Fused dual GEMM: out[m,n] = sum_k A1[m,k]*Wrel[n,k] + sum_k A2[m,k]*Wroot[n,k] + bias[n]
// optionally ReLU'd. M parallel over waves; each wave computes a 16x64 tile
// with V_WMMA_F32_16X16X4_F32 (K-loop of 16 steps, 4 N-tiles, 2 matmuls).
//
// Fragment layouts (wave32, ISA 7.12.2):
//   A 16x4 f32 : lane L holds A[L%16][k0 + 2*(L>>4) + {0,1}]   -> float2 load
//   B 4x16 f32 : lane L holds B[k0 + 2*(L>>4) + {0,1}][L%16]
//                with B[k][n] = W[n][k]                          -> float2 from W row
//   C 16x16 f32: VGPR r, lane L -> row r + 8*(L>>4), col L%16
// ---------------------------------------------------------------------------
__global__ __launch_bounds__(256) void gemm_dual_kernel(
    const float* __restrict__ A1,     // [M,64]  aggregated neighbors
    const float* __restrict__ A2,     // [M,64]  root features
    const float* __restrict__ Wrel,   // [64,64] row-major (out,in)
    const float* __restrict__ Wroot,  // [64,64]
    const float* __restrict__ bias,   // [64]
    float* __restrict__ out,          // [M,64]
    int M, int doRelu) {
    __shared__ float sW[2][DIM][DIM + 2];   // +2 pad: kill LDS bank conflicts

    int t = threadIdx.x;
    // Preload both weight matrices into LDS.
    for (int i = t; i < DIM * DIM; i += 256) {
        int r = i >> 6, c = i & 63;
        sW[0][r][c] = Wrel[i];
        sW[1][r][c] = Wroot[i];
    }
    __syncthreads();

    int wave = t >> 5;
    int lane = t & 31;
    int lo = lane & 15;        // M (A) / N (B,C) within tile
    int hi = lane >> 4;        // K-pair select / C row-half select

    int mtile = blockIdx.x * 8 + wave;
    if (mtile * 16 >= M) return;           // wave-uniform: EXEC stays all-1s
    int m0 = mtile * 16;

    const float* a1p = A1 + (size_t)(m0 + lo) * DIM;
    const float* a2p = A2 + (size_t)(m0 + lo) * DIM;

    v8f acc[4] = {};   // 4 N-tiles of 16x16 f32

    #pragma unroll
    for (int ks = 0; ks < 16; ++ks) {
        int k0 = ks * 4 + 2 * hi;
        v2f a1 = *(const v2f*)(a1p + k0);
        v2f a2 = *(const v2f*)(a2p + k0);
        #pragma unroll
        for (int n = 0; n < 4; ++n) {
            v2f b1 = *(const v2f*)(&sW[0][n * 16 + lo][k0]);
            v2f b2 = *(const v2f*)(&sW[1][n * 16 + lo][k0]);
            acc[n] = __builtin_amdgcn_wmma_f32_16x16x4_f32(
                false, a1, false, b1, (short)0, acc[n], false, false);
            acc[n] = __builtin_amdgcn_wmma_f32_16x16x4_f32(
                false, a2, false, b2, (short)0, acc[n], false, false);
        }
    }

    // Bias (+ ReLU) and store. C layout: acc[n][r] -> row m0 + r + 8*hi,
    // col n*16 + lo.
    #pragma unroll
    for (int n = 0; n < 4; ++n) {
        float bl = bias[n * 16 + lo];
        #pragma unroll
        for (int r = 0; r < 8; ++r) {
            float v = acc[n][r] + bl;
            if (doRelu) v = fmaxf(v, 0.0f);
            out[(size_t)(m0 + r + 8 * hi) * DIM + n * 16 + lo] = v;
        }
    }
}

// ---------------------------------------------------------------------------
// Launch: zero -> scatter(x) -> GEMM+ReLU -> zero -> scatter(h) -> GEMM.
// ---------------------------------------------------------------------------
extern "C" void kernel_launch(void* const* d_in, const int* in_sizes, int n_in,
                              void* d_out, int out_size, void* d_ws, size_t ws_size,
                              hipStream_t stream) {
    const float* x       = (const float*)d_in[0];
    const int*   ei      = (const int*)  d_in[1];   // [2, N_EDGES]
    const float* W1_rel  = (const float*)d_in[2];
    const float* b1      = (const float*)d_in[3];
    const float* W1_root = (const float*)d_in[4];
    const float* W2_rel  = (const float*)d_in[5];
    const float* b2      = (const float*)d_in[6];
    const float* W2_root = (const float*)d_in[7];
    float* out = (float*)d_out;

    const int* src = ei;
    const int* dst = ei + N_EDGES;

    float* agg = (float*)d_ws;                       // [N_NODES, 64]
    float* h   = agg + (size_t)N_NODES * DIM;        // [N_NODES, 64]

    const int featN4    = (N_NODES * DIM) / 4;       // float4 count
    const int zeroBlks  = (featN4 + 255) / 256;
    const long long scatterThreads = (long long)N_EDGES * 16;
    const int scatterBlks = (int)((scatterThreads + 255) / 256);
    const int mtiles    = (N_NODES + 15) / 16;       // 6250
    const int gemmBlks  = (mtiles + 7) / 8;          // 782

    // Layer 1: h = relu(agg(x) @ W1_rel.T + x @ W1_root.T + b1)
    zero_kernel<<<zeroBlks, 256, 0, stream>>>(agg, featN4);
    scatter_add_kernel<<<scatterBlks, 256, 0, stream>>>(x, src, dst, agg);
    gemm_dual_kernel<<<gemmBlks, 256, 0, stream>>>(agg, x, W1_rel, W1_root, b1,
                                                   h, N_NODES, 1);

    // Layer 2: out = agg(h) @ W2_rel.T + h @ W2_root.T + b2
    zero_kernel<<<zeroBlks, 256, 0, stream>>>(agg, featN4);
    scatter_add_kernel<<<scatterBlks, 256, 0, stream>>>(h, src, dst, agg);
    gemm_dual_kernel<<<gemmBlks, 256, 0, stream>>>(agg, h, W2_rel, W2_root, b2,
                                                   out, N_NODES, 0);
}